// HypersphericalLoss_41266045780102
// MI455X (gfx1250) — compile-verified
//
#include <hip/hip_runtime.h>
#include <hip/hip_bf16.h>

// HypersphericalLoss on MI455X (gfx1250, wave32).
//
// loss = 1 - mean(n @ n^T) with n = row-L2-normalized embeddings [N=16384, D=128].
// Algebraic identity: sum_{i,j} n_i . n_j = || sum_i n_i ||^2, so
//   loss = 1 - ||s||^2 / N^2,  s = sum_i e_i / max(||e_i||, 1e-12)  (s in R^128)
// => one streaming pass over 8 MB (HBM floor ~0.36us @ 23.3 TB/s), no N^2 GEMM.

#define N_DIM 128
#define BLOCKS 128
#define TPB 256
#define WAVES_PER_BLOCK (TPB / 32)
#define TPB2 256
#define WAVES2 (TPB2 / 32)

typedef __attribute__((ext_vector_type(2))) float v2f;
typedef __attribute__((ext_vector_type(8))) float v8f;

// Kernel 1: per-block partial of s = sum of normalized rows.
// One wave per row-slice: 32 lanes x float4 = 128 contiguous floats (512B) per row.
__global__ __launch_bounds__(TPB) void hs_rowsum_kernel(
    const float* __restrict__ emb, float* __restrict__ partials, int nrows) {
  __shared__ float lds[WAVES_PER_BLOCK][N_DIM];

  const int lane   = threadIdx.x & 31;
  const int wave   = threadIdx.x >> 5;
  const int gwave  = blockIdx.x * WAVES_PER_BLOCK + wave;
  const int wstride = gridDim.x * WAVES_PER_BLOCK;

  float a0 = 0.f, a1 = 0.f, a2 = 0.f, a3 = 0.f;

  for (int row = gwave; row < nrows; row += wstride) {
    const float4 v = ((const float4*)(emb + (size_t)row * N_DIM))[lane];

    // Prefetch the next row this wave will touch (emits global_prefetch_b8).
    if (row + wstride < nrows)
      __builtin_prefetch(emb + (size_t)(row + wstride) * N_DIM + lane * 4, 0, 0);

    // ||row||^2 via per-lane partial + wave32 xor-shuffle tree.
    float p = v.x * v.x + v.y * v.y + v.z * v.z + v.w * v.w;
#pragma unroll
    for (int off = 16; off > 0; off >>= 1) p += __shfl_xor(p, off, 32);

    const float scale = 1.0f / fmaxf(sqrtf(p), 1e-12f);  // matches F.normalize eps
    a0 = fmaf(v.x, scale, a0);
    a1 = fmaf(v.y, scale, a1);
    a2 = fmaf(v.z, scale, a2);
    a3 = fmaf(v.w, scale, a3);
  }

  lds[wave][lane * 4 + 0] = a0;
  lds[wave][lane * 4 + 1] = a1;
  lds[wave][lane * 4 + 2] = a2;
  lds[wave][lane * 4 + 3] = a3;
  __syncthreads();

  // Deterministic per-block combine (fixed order, no fp atomics).
  for (int t = threadIdx.x; t < N_DIM; t += TPB) {
    float s = 0.f;
#pragma unroll
    for (int w = 0; w < WAVES_PER_BLOCK; ++w) s += lds[w][t];
    partials[(size_t)blockIdx.x * N_DIM + t] = s;
  }
}

// Kernel 2: s = sum of block partials; ||s||^2 via V_WMMA_F32_16X16X4_F32 with
// A = ones(16x4), B = squared per-lane partials, C = 0.  Because A==1 and C==0,
// sum(all 256 entries of D) == 16 * sum(all entries of B) == 16 * ||s||^2,
// which is invariant to the exact WMMA VGPR slot->element mapping.
__global__ __launch_bounds__(TPB2) void hs_final_kernel(
    const float* __restrict__ partials, float* __restrict__ out,
    int nblocks, float inv_n2) {
  __shared__ float lds[WAVES2][N_DIM];

  const int lane = threadIdx.x & 31;
  const int wave = threadIdx.x >> 5;

  float a0 = 0.f, a1 = 0.f, a2 = 0.f, a3 = 0.f;
  for (int b = wave; b < nblocks; b += WAVES2) {
    const float4 v = ((const float4*)(partials + (size_t)b * N_DIM))[lane];
    a0 += v.x; a1 += v.y; a2 += v.z; a3 += v.w;
  }
  lds[wave][lane * 4 + 0] = a0;
  lds[wave][lane * 4 + 1] = a1;
  lds[wave][lane * 4 + 2] = a2;
  lds[wave][lane * 4 + 3] = a3;
  __syncthreads();

  if (wave == 0) {  // full wave32 active -> EXEC all ones for WMMA
    float s0 = 0.f, s1 = 0.f, s2 = 0.f, s3 = 0.f;
#pragma unroll
    for (int w = 0; w < WAVES2; ++w) {
      s0 += lds[w][lane * 4 + 0];
      s1 += lds[w][lane * 4 + 1];
      s2 += lds[w][lane * 4 + 2];
      s3 += lds[w][lane * 4 + 3];
    }
    // Two squared partials per lane: 64 slots total == the whole of ||s||^2.
    v2f A; A[0] = 1.0f; A[1] = 1.0f;
    v2f B; B[0] = s0 * s0 + s1 * s1;
           B[1] = s2 * s2 + s3 * s3;
    v8f C = {};
    // 8 args: (neg_a, A, neg_b, B, c_mod, C, reuse_a, reuse_b)
    v8f D = __builtin_amdgcn_wmma_f32_16x16x4_f32(
        false, A, false, B, (short)0, C, false, false);

    // Layout-independent total: sum every D entry over the wave -> 16*||s||^2.
    float t = D[0] + D[1] + D[2] + D[3] + D[4] + D[5] + D[6] + D[7];
#pragma unroll
    for (int off = 16; off > 0; off >>= 1) t += __shfl_xor(t, off, 32);

    if (lane == 0) out[0] = 1.0f - (t * (1.0f / 16.0f)) * inv_n2;
  }
}

extern "C" void kernel_launch(void* const* d_in, const int* in_sizes, int n_in,
                              void* d_out, int out_size, void* d_ws, size_t ws_size,
                              hipStream_t stream) {
  const float* emb = (const float*)d_in[0];
  const int nrows = in_sizes[0] / N_DIM;  // 16384

  float* partials = (float*)d_ws;  // BLOCKS * N_DIM * 4 = 64 KB
  float* out = (float*)d_out;

  const float inv_n2 = 1.0f / ((float)nrows * (float)nrows);

  hipLaunchKernelGGL(hs_rowsum_kernel, dim3(BLOCKS), dim3(TPB), 0, stream,
                     emb, partials, nrows);
  hipLaunchKernelGGL(hs_final_kernel, dim3(1), dim3(TPB2), 0, stream,
                     partials, out, BLOCKS, inv_n2);
}